// RoIPool_12713103196737
// MI455X (gfx1250) — compile-verified
//
#include <hip/hip_runtime.h>
#include <cstdint>

// RoIPool for MI455X (gfx1250):
//  - memory-bound; features (105 MB) are L2-resident (192 MB L2)
//  - Tensor Data Mover stages RoI windows (8-ch x th x tw strided 3-D tiles)
//    into LDS, double-buffered: TDM for tile t+1 runs while waves compute
//    tile t (s_wait_tensorcnt 1 / 0 pipeline)
//  - outputs streamed with non-temporal stores (write-once, never re-read)
//  - wave32: 256 threads = 8 waves per block

static constexpr int PHB = 7, PWB = 7;
static constexpr int Cc = 256, Hh = 160, Ww = 160;
static constexpr int CT = 8;       // channels per TDM tile
static constexpr int NTILES = 4;   // tiles per block -> 32 channels per block
static constexpr int MAXT = 40;    // max RoI tile extent (<= 39+1 from setup bounds)
static constexpr int TILE_ELEMS = CT * MAXT * MAXT;   // 12800 floats = 51.2 KB
static constexpr float NEGV = -1e30f;

typedef unsigned int u32x4 __attribute__((ext_vector_type(4)));
typedef int i32x8 __attribute__((ext_vector_type(8)));
typedef int i32x4 __attribute__((ext_vector_type(4)));

__global__ __launch_bounds__(256) void roipool_tdm(
    const float* __restrict__ feat, const float* __restrict__ rois,
    float* __restrict__ out, int N)
{
    __shared__ float smem[2][TILE_ELEMS];   // 102.4 KB, double-buffered

    const int bid = blockIdx.x;
    const int roi   = bid >> 3;                  // 8 blocks per roi
    const int cbase = (bid & 7) * (CT * NTILES); // 32 channels per block
    if (roi >= N) return;

    // ---- per-RoI geometry (block-uniform, scalarizes to SALU; bit-exact) ----
    const float* r = rois + (size_t)roi * 5;
    const int b  = (int)r[0];
    const int x1 = (int)rintf(r[1] * 0.0625f);   // jnp.round = half-to-even = rintf
    const int y1 = (int)rintf(r[2] * 0.0625f);
    const int x2 = (int)rintf(r[3] * 0.0625f);
    const int y2 = (int)rintf(r[4] * 0.0625f);
    const float bsw = (float)max(x2 - x1 + 1, 1) / (float)PWB;
    const float bsh = (float)max(y2 - y1 + 1, 1) / (float)PHB;

    // Tile bounds: bin starts/ends are monotonic -> tile = [hs0,he6) x [ws0,we6)
    const int ws0 = min(max(x1, 0), Ww);
    const int hs0 = min(max(y1, 0), Hh);
    const int we6 = min(max((int)ceilf(7.0f * bsw) + x1, 0), Ww);
    const int he6 = min(max((int)ceilf(7.0f * bsh) + y1, 0), Hh);
    const int tw = min(max(we6 - ws0, 1), MAXT);
    const int th = min(max(he6 - hs0, 1), MAXT);

    const uint64_t roi_gbase = (uint64_t)(uintptr_t)feat +
        4ull * ((uint64_t)(b * Cc + cbase) * (uint64_t)(Hh * Ww)
                + (uint64_t)hs0 * (uint64_t)Ww + (uint64_t)ws0);
    const uint32_t lds0 = (uint32_t)(uintptr_t)&smem[0][0]; // low 32 bits = LDS offset

#if __has_builtin(__builtin_amdgcn_tensor_load_to_lds) && __has_builtin(__builtin_amdgcn_s_wait_tensorcnt)
    // Issue the TDM load of channel-tile t into buffer (t&1). Wave-level op.
    auto tdm_issue = [&](int t) {
        const uint64_t gaddr = roi_gbase + 4ull * (uint64_t)(t * CT) * (uint64_t)(Hh * Ww);
        u32x4 g0;
        g0[0] = 1u;                                        // count=1 valid descriptor
        g0[1] = lds0 + (uint32_t)((t & 1) * TILE_ELEMS * 4); // lds_addr (bytes)
        g0[2] = (uint32_t)(gaddr & 0xFFFFFFFFu);           // global_addr[31:0]
        g0[3] = (uint32_t)((gaddr >> 32) & 0x01FFFFFFu)    // global_addr[56:32]
                | 0x80000000u;                             // type=2 ("image")
        i32x8 g1;
        g1[0] = 2 << 16;                      // data_size = 4 bytes; no mcast/pad/iter
        g1[1] = (tw & 0xFFFF) << 16;          // tensor_dim0[15:0]  (bits 63:48)
        g1[2] = (th & 0xFFFF) << 16;          // tensor_dim0 hi=0 | tensor_dim1[15:0]
        g1[3] = (tw & 0xFFFF) << 16;          // tensor_dim1 hi=0 | tile_dim0 = tw
        g1[4] = (th & 0xFFFF) | (CT << 16);   // tile_dim1 = th | tile_dim2 = CT
        g1[5] = Ww;                           // tensor_dim0_stride[31:0] = 160
        g1[6] = (Hh * Ww) << 16;              // d0s[47:32]=0 | tensor_dim1_stride[15:0]
        g1[7] = 0;                            // tensor_dim1_stride[47:16] = 0
        i32x4 g2; g2[0] = CT; g2[1] = 0; g2[2] = 0; g2[3] = 0; // tensor_dim2=CT (3-D)
        i32x4 g3 = {0, 0, 0, 0};
#if __clang_major__ >= 23
        i32x8 g4 = {0, 0, 0, 0, 0, 0, 0, 0};  // clang-23 toolchain: 6-arg builtin
        __builtin_amdgcn_tensor_load_to_lds(g0, g1, g2, g3, g4, 0);
#else
        __builtin_amdgcn_tensor_load_to_lds(g0, g1, g2, g3, 0);
#endif
    };

    if (threadIdx.x < 32) tdm_issue(0);   // prologue: prefetch tile 0
#else
    auto coop_load = [&](int t) {         // fallback: cooperative staging
        float* dst = &smem[t & 1][0];
        for (int i = threadIdx.x; i < CT * th * tw; i += 256) {
            const int cl = i / (th * tw), rem = i % (th * tw);
            const int y = rem / tw, x = rem - y * tw;
            dst[i] = feat[(size_t)((b * Cc + cbase + t * CT + cl) * Hh + hs0 + y) * Ww
                          + ws0 + x];
        }
    };
    coop_load(0);
#endif

    for (int t = 0; t < NTILES; ++t) {
#if __has_builtin(__builtin_amdgcn_tensor_load_to_lds) && __has_builtin(__builtin_amdgcn_s_wait_tensorcnt)
        if (threadIdx.x < 32) {
            if (t + 1 < NTILES) {
                tdm_issue(t + 1);                                // overlap next DMA
                __builtin_amdgcn_s_wait_tensorcnt((unsigned short)1); // tile t done
            } else {
                __builtin_amdgcn_s_wait_tensorcnt((unsigned short)0);
            }
        }
        __syncthreads();   // tile t visible to all 8 waves
#else
        if (t + 1 < NTILES) coop_load(t + 1);
        __syncthreads();
#endif

        // ---- 49 bins x CT channels for this tile, max-reduce from LDS ----
        const float* buf = &smem[t & 1][0];
        const size_t obase = ((size_t)roi * Cc + cbase + t * CT) * (size_t)(PHB * PWB);
        for (int o = threadIdx.x; o < CT * PHB * PWB; o += 256) {
            const int cl  = o / (PHB * PWB);
            const int bin = o - cl * (PHB * PWB);
            const int ph  = bin / PWB;
            const int pw  = bin - ph * PWB;
            const int hs = min(max((int)floorf((float)ph * bsh) + y1, 0), Hh);
            const int he = min(max((int)ceilf((float)(ph + 1) * bsh) + y1, 0), Hh);
            const int ws = min(max((int)floorf((float)pw * bsw) + x1, 0), Ww);
            const int we = min(max((int)ceilf((float)(pw + 1) * bsw) + x1, 0), Ww);

            float v = 0.0f;  // invalid bins -> 0 (matches reference)
            if (he > hs && we > ws) {
                float acc = NEGV;
                const float* tile = buf + cl * (th * tw);
                for (int h = hs; h < he; ++h) {
                    const float* row = tile + (h - hs0) * tw - ws0;
                    for (int w = ws; w < we; ++w)
                        acc = fmaxf(acc, row[w]);
                }
                v = acc;
            }
            // write-once output: NT store keeps L2 for the feature map
            __builtin_nontemporal_store(v, out + obase + (size_t)o);
        }
        __syncthreads();   // buffer (t&1) free before load t+2 overwrites it
    }
}

extern "C" void kernel_launch(void* const* d_in, const int* in_sizes, int n_in,
                              void* d_out, int out_size, void* d_ws, size_t ws_size,
                              hipStream_t stream) {
    const float* feat = (const float*)d_in[0];
    const float* rois = (const float*)d_in[1];
    float* out = (float*)d_out;
    const int N = in_sizes[1] / 5;                    // 1000 RoIs
    const int blocks = N * (Cc / (CT * NTILES));      // 8,000 blocks
    roipool_tdm<<<dim3(blocks), dim3(256), 0, stream>>>(feat, rois, out, N);
}